// TextGNN_58884001628164
// MI455X (gfx1250) — compile-verified
//
#include <hip/hip_runtime.h>
#include <hip/hip_bf16.h>
#include <math.h>

// GAT layer for MI455X (gfx1250, wave32).
// Phases: fp32 WMMA GEMM -> per-node logits -> segment softmax (int-encoded
// atomic max) -> weighted scatter-add aggregation (f32 global atomics).

typedef __attribute__((ext_vector_type(2))) float v2f;
typedef __attribute__((ext_vector_type(8))) float v8f;

#define NEG_SLOPE 0.2f
#define EPSV 1e-16f

// ---------- order-preserving float <-> int encoding for atomicMax ----------
__device__ __forceinline__ int enc_f(float f) {
    int i = __float_as_int(f);
    return i >= 0 ? i : (i ^ 0x7FFFFFFF);
}
__device__ __forceinline__ float dec_f(int i) {
    return __int_as_float(i >= 0 ? i : (i ^ 0x7FFFFFFF));
}
#define ENC_NEG_INF 0x807FFFFF  // enc_f(-INFINITY)

// =====================================================================
// Kernel 1: xw = x @ W   (N x 128) = (N x 128)(128 x 128), fp32 WMMA.
// One wave per 16-row slab; 8 column tiles of 16; K in chunks of 4.
// Weight pre-swizzled into LDS in the exact B-fragment layout so each
// fragment is one ds_load_b64 per lane.
//   A frag (16x4 f32): lane l -> m = base+(l&15), k = 4*kk + 2*(l>>4) + {0,1}
//   B frag (4x16 f32): lane l -> n = 16*ct+(l&15), k = 4*kk + 2*(l>>4) + {0,1}
//   C/D  (16x16 f32):  lane l, vgpr v -> n = 16*ct+(l&15), m = base+8*(l>>4)+v
// =====================================================================
__global__ __launch_bounds__(256) void gat_gemm_xw(const float* __restrict__ x,
                                                   const float* __restrict__ w,
                                                   float* __restrict__ xw,
                                                   int N) {
    __shared__ float wlds[128 * 128];  // 64 KB of the WGP's 320 KB LDS

    const int tid = threadIdx.x;
    // Cooperative fill: linear LDS index i -> (tile, lane, half) -> (k, n)
    for (int i = tid; i < 128 * 128; i += 256) {
        int t    = i >> 6;        // tile = kk*8 + ct
        int rem  = i & 63;
        int l    = rem >> 1;      // lane within wave
        int half = rem & 1;
        int kk = t >> 3;
        int ct = t & 7;
        int n  = ct * 16 + (l & 15);
        int k  = kk * 4 + ((l >> 4) << 1) + half;
        wlds[i] = w[k * 128 + n];
    }
    __syncthreads();

    const int wave = tid >> 5;
    const int lane = tid & 31;
    const int rowBase = (blockIdx.x * 8 + wave) * 16;
    if (rowBase >= N) return;  // wave-uniform: EXEC stays all-ones for WMMA

    const int mRow = rowBase + (lane & 15);
    const int mClamped = mRow < N ? mRow : N - 1;
    const int kOff = (lane >> 4) << 1;  // 0 or 2

    v8f acc[8];
#pragma unroll
    for (int ct = 0; ct < 8; ++ct) acc[ct] = (v8f){0.f,0.f,0.f,0.f,0.f,0.f,0.f,0.f};

    for (int kk = 0; kk < 32; ++kk) {
        const float* ap = x + (size_t)mClamped * 128 + kk * 4 + kOff;
        v2f a;
        a.x = ap[0];
        a.y = ap[1];
#pragma unroll
        for (int ct = 0; ct < 8; ++ct) {
            const float* bp = &wlds[(kk * 8 + ct) * 64 + lane * 2];
            v2f b;
            b.x = bp[0];
            b.y = bp[1];
            // v_wmma_f32_16x16x4_f32
            acc[ct] = __builtin_amdgcn_wmma_f32_16x16x4_f32(
                false, a, false, b, (short)0, acc[ct], false, false);
        }
    }

    const int rOut = rowBase + ((lane >> 4) << 3);
    const int cOut = (lane & 15);
#pragma unroll
    for (int ct = 0; ct < 8; ++ct) {
#pragma unroll
        for (int v = 0; v < 8; ++v) {
            int r = rOut + v;
            if (r < N) xw[(size_t)r * 128 + ct * 16 + cOut] = acc[ct][v];
        }
    }
}

// =====================================================================
// Kernel 2: a_i[n][h] = sum_c xw[n][h][c]*att[h][c]
//           a_j[n][h] = sum_c xw[n][h][c]*att[h][C+c]
// One block per node, 4 waves = 4 heads, shuffle reduction per wave32.
// =====================================================================
__global__ __launch_bounds__(128) void gat_compute_a(const float* __restrict__ xw,
                                                     const float* __restrict__ att,
                                                     float* __restrict__ ai,
                                                     float* __restrict__ aj,
                                                     int N) {
    const int node = blockIdx.x;
    const int tid  = threadIdx.x;
    const int h = tid >> 5;
    const int c = tid & 31;
    float v  = xw[(size_t)node * 128 + tid];
    float pi = v * att[h * 64 + c];
    float pj = v * att[h * 64 + 32 + c];
#pragma unroll
    for (int off = 16; off > 0; off >>= 1) {
        pi += __shfl_down(pi, off, 32);
        pj += __shfl_down(pj, off, 32);
    }
    if (c == 0) {
        ai[node * 4 + h] = pi;
        aj[node * 4 + h] = pj;
    }
}

// =====================================================================
// Kernel 3: init out = bias (broadcast), m = enc(-inf), denom = 0
// =====================================================================
__global__ void gat_init(float* __restrict__ out, const float* __restrict__ bias,
                         int* __restrict__ m_enc, float* __restrict__ denom, int N) {
    int i = blockIdx.x * blockDim.x + threadIdx.x;
    if (i < N * 128) out[i] = bias[i & 127];
    if (i < N * 4) {
        m_enc[i] = ENC_NEG_INF;
        denom[i] = 0.f;
    }
}

// =====================================================================
// Kernel 4: per (edge, head): alpha = leaky_relu(a_i[row]+a_j[col]);
//           store alpha; segment max via int-encoded atomicMax.
// =====================================================================
__global__ void gat_alpha_max(const int* __restrict__ ei,
                              const float* __restrict__ ai,
                              const float* __restrict__ aj,
                              float* __restrict__ alpha_ws,
                              int* __restrict__ m_enc,
                              int E, int N) {
    int idx = blockIdx.x * blockDim.x + threadIdx.x;
    int Etot = E + N;
    if (idx >= Etot * 4) return;
    int e = idx >> 2;
    int h = idx & 3;
    int r, c;
    if (e < E) { r = ei[e]; c = ei[E + e]; }
    else       { r = c = e - E; }  // self loop
    float a = ai[r * 4 + h] + aj[c * 4 + h];
    a = a >= 0.f ? a : NEG_SLOPE * a;
    alpha_ws[idx] = a;  // idx == e*4 + h
    atomicMax(&m_enc[r * 4 + h], enc_f(a));
}

// =====================================================================
// Kernel 5: e = exp(alpha - m[row]); denom[row] += e (atomic); e -> ws
// =====================================================================
__global__ void gat_exp_sum(const int* __restrict__ ei,
                            const int* __restrict__ m_enc,
                            float* __restrict__ e_ws,
                            float* __restrict__ denom,
                            int E, int N) {
    int idx = blockIdx.x * blockDim.x + threadIdx.x;
    int Etot = E + N;
    if (idx >= Etot * 4) return;
    int e = idx >> 2;
    int h = idx & 3;
    int r;
    if (e < E) r = ei[e];
    else       r = e - E;
    float a  = e_ws[idx];
    float mv = dec_f(m_enc[r * 4 + h]);
    float ex = expf(a - mv);
    e_ws[idx] = ex;
    atomicAdd(&denom[r * 4 + h], ex);
}

// =====================================================================
// Kernel 6: out[row][hc] += xw[col][hc] * e/(denom[row]+eps)
// One 128-thread block per edge; thread = output column; f32 global atomics.
// Both xw (25.6MB) and out (25.6MB) are L2-resident on MI455X (192MB L2).
// =====================================================================
__global__ __launch_bounds__(128) void gat_aggregate(const int* __restrict__ ei,
                                                     const float* __restrict__ xw,
                                                     const float* __restrict__ e_ws,
                                                     const float* __restrict__ denom,
                                                     float* __restrict__ out,
                                                     int E, int N) {
    int e   = blockIdx.x;
    int tid = threadIdx.x;
    int h   = tid >> 5;
    int r, c;
    if (e < E) { r = ei[e]; c = ei[E + e]; }
    else       { r = c = e - E; }
    float coef = e_ws[e * 4 + h] / (denom[r * 4 + h] + EPSV);
    float val  = xw[(size_t)c * 128 + tid] * coef;
    atomicAdd(&out[(size_t)r * 128 + tid], val);
}

// =====================================================================
extern "C" void kernel_launch(void* const* d_in, const int* in_sizes, int n_in,
                              void* d_out, int out_size, void* d_ws, size_t ws_size,
                              hipStream_t stream) {
    const float* x    = (const float*)d_in[0];
    const int*   ei   = (const int*)d_in[1];    // [2, E] flat
    const float* w    = (const float*)d_in[2];  // [128, 128]
    const float* att  = (const float*)d_in[3];  // [1, 4, 64] flat
    const float* bias = (const float*)d_in[4];  // [128]
    float* out = (float*)d_out;

    const int N = in_sizes[0] / 128;
    const int E = in_sizes[1] / 2;
    const int Etot = E + N;

    // workspace layout (floats): xw | a_i | a_j | m_enc(int) | denom | alpha/e
    float* ws    = (float*)d_ws;
    float* xw    = ws;
    float* ai    = xw + (size_t)N * 128;
    float* aj    = ai + (size_t)N * 4;
    int*   menc  = (int*)(aj + (size_t)N * 4);
    float* denom = (float*)(menc + (size_t)N * 4);
    float* alpha = denom + (size_t)N * 4;

    // 1) GEMM: 8 waves/block, 16 rows/wave -> 128 rows/block
    int gemmBlocks = (N + 127) / 128;
    gat_gemm_xw<<<gemmBlocks, 256, 0, stream>>>(x, w, xw, N);

    // 2) per-node attention logits
    gat_compute_a<<<N, 128, 0, stream>>>(xw, att, ai, aj, N);

    // 3) init out=bias, m=-inf, denom=0
    int initBlocks = (N * 128 + 255) / 256;
    gat_init<<<initBlocks, 256, 0, stream>>>(out, bias, menc, denom, N);

    // 4) leaky-relu logits + segment max
    int ehBlocks = (Etot * 4 + 255) / 256;
    gat_alpha_max<<<ehBlocks, 256, 0, stream>>>(ei, ai, aj, alpha, menc, E, N);

    // 5) exp + segment sum
    gat_exp_sum<<<ehBlocks, 256, 0, stream>>>(ei, menc, alpha, denom, E, N);

    // 6) weighted scatter-add aggregation
    gat_aggregate<<<Etot, 128, 0, stream>>>(ei, xw, alpha, denom, out, E, N);
}